// DiffTreeMachine_24120536334589
// MI455X (gfx1250) — compile-verified
//
#include <hip/hip_runtime.h>
#include <math.h>

#define Bdim 32
#define L0c  24
#define NSTEPS 8
#define Lc   32
#define Fc   64
#define Rc   512
#define FRc  (Fc*Rc)        /* 32768 */
#define DMc  512
#define DFFc 2048
#define NHc  8
#define DHc  64
#define Tc   (Lc+2)         /* 34 tokens */

typedef __attribute__((ext_vector_type(2))) float v2f;
typedef __attribute__((ext_vector_type(8))) float v8f;

__device__ __forceinline__ v8f wmma4(v2f a, v2f b, v8f c) {
    // D = A(16x4,f32) * B(4x16,f32) + C(16x16,f32), exact f32 WMMA
    return __builtin_amdgcn_wmma_f32_16x16x4_f32(false, a, false, b, (short)0, c, false, false);
}

__device__ __forceinline__ float gelu_exact(float x) {
    return 0.5f * x * (1.0f + erff(x * 0.70710678118654752f));
}

// ---------------------------------------------------------------------------
// Generic WMMA GEMM: Y[M,N] = act(X[M,K] @ W[N,K]^T + bias) (+ residual)
// One wave computes a 16x64 tile of Y; inner loop is 2-stage software
// pipelined so global loads for future k-steps overlap current WMMAs.
// M%16==0, N%64==0, K%8==0 guaranteed by caller.
// ---------------------------------------------------------------------------
__global__ void gemm_xwT_kernel(const float* __restrict__ X, const float* __restrict__ W,
                                const float* __restrict__ bias, const float* __restrict__ residual,
                                float* __restrict__ Y, int M, int N, int K, int act)
{
    int lane = threadIdx.x & 31;
    int wave = blockIdx.x * (blockDim.x >> 5) + (threadIdx.x >> 5);
    int tilesN = N >> 6;
    int mT = wave / tilesN;
    int n64 = wave - mT * tilesN;
    if (mT * 16 >= M) return;                       // wave-uniform
    int lo = lane & 15, hi = lane >> 4, k2 = hi * 2;

    const float* Arow = X + (size_t)(mT * 16 + lo) * K + k2;
    const float* B0 = W + (size_t)(n64 * 64 + lo) * K + k2;
    const float* B1 = B0 + (size_t)16 * K;
    const float* B2 = B0 + (size_t)32 * K;
    const float* B3 = B0 + (size_t)48 * K;

    v8f c0 = (v8f)0.0f, c1 = (v8f)0.0f, c2 = (v8f)0.0f, c3 = (v8f)0.0f;

    // stage 0 (k=0) and stage 1 (k=4) preload
    v2f a0 = *(const v2f*)(Arow);
    v2f p0 = *(const v2f*)(B0);
    v2f p1 = *(const v2f*)(B1);
    v2f p2 = *(const v2f*)(B2);
    v2f p3 = *(const v2f*)(B3);
    v2f a1 = *(const v2f*)(Arow + 4);
    v2f q0 = *(const v2f*)(B0 + 4);
    v2f q1 = *(const v2f*)(B1 + 4);
    v2f q2 = *(const v2f*)(B2 + 4);
    v2f q3 = *(const v2f*)(B3 + 4);

    for (int k = 8; k + 8 <= K; k += 8) {
        // issue loads for k-step k, then consume stage 0
        v2f na  = *(const v2f*)(Arow + k);
        v2f nb0 = *(const v2f*)(B0 + k);
        v2f nb1 = *(const v2f*)(B1 + k);
        v2f nb2 = *(const v2f*)(B2 + k);
        v2f nb3 = *(const v2f*)(B3 + k);
        c0 = wmma4(a0, p0, c0);
        c1 = wmma4(a0, p1, c1);
        c2 = wmma4(a0, p2, c2);
        c3 = wmma4(a0, p3, c3);
        a0 = na; p0 = nb0; p1 = nb1; p2 = nb2; p3 = nb3;
        // issue loads for k-step k+4, then consume stage 1
        v2f ma  = *(const v2f*)(Arow + k + 4);
        v2f mb0 = *(const v2f*)(B0 + k + 4);
        v2f mb1 = *(const v2f*)(B1 + k + 4);
        v2f mb2 = *(const v2f*)(B2 + k + 4);
        v2f mb3 = *(const v2f*)(B3 + k + 4);
        c0 = wmma4(a1, q0, c0);
        c1 = wmma4(a1, q1, c1);
        c2 = wmma4(a1, q2, c2);
        c3 = wmma4(a1, q3, c3);
        a1 = ma; q0 = mb0; q1 = mb1; q2 = mb2; q3 = mb3;
    }
    // drain the two remaining stages
    c0 = wmma4(a0, p0, c0); c1 = wmma4(a0, p1, c1);
    c2 = wmma4(a0, p2, c2); c3 = wmma4(a0, p3, c3);
    c0 = wmma4(a1, q0, c0); c1 = wmma4(a1, q1, c1);
    c2 = wmma4(a1, q2, c2); c3 = wmma4(a1, q3, c3);

#pragma unroll
    for (int i = 0; i < 8; ++i) {
        int m = mT * 16 + hi * 8 + i;
        size_t rowOff = (size_t)m * N;
        int col = n64 * 64 + lo;
        float v0 = c0[i] + bias[col];
        float v1 = c1[i] + bias[col + 16];
        float v2 = c2[i] + bias[col + 32];
        float v3 = c3[i] + bias[col + 48];
        if (act) { v0 = gelu_exact(v0); v1 = gelu_exact(v1); v2 = gelu_exact(v2); v3 = gelu_exact(v3); }
        if (residual) {
            v0 += residual[rowOff + col];
            v1 += residual[rowOff + col + 16];
            v2 += residual[rowOff + col + 32];
            v3 += residual[rowOff + col + 48];
        }
        Y[rowOff + col]      = v0;
        Y[rowOff + col + 16] = v1;
        Y[rowOff + col + 32] = v2;
        Y[rowOff + col + 48] = v3;
    }
}

// ---------------------------------------------------------------------------
// Incremental ctrl-net encoder: enc[b,l,:] = mem[b,l,:,:].flat @ ctrl_W.T + b
// Encodes slots [lBegin, lBegin+lCount) for all batches. K = 32768.
// Same 2-stage pipelined WMMA loop + L2 prefetch hints.
// ---------------------------------------------------------------------------
__global__ void encode_slots_kernel(const float* __restrict__ mem, const float* __restrict__ ctrlW,
                                    const float* __restrict__ ctrlb, float* __restrict__ encSlots,
                                    int lBegin, int lCount)
{
    const int K = FRc;
    int lane = threadIdx.x & 31;
    int wave = blockIdx.x * (blockDim.x >> 5) + (threadIdx.x >> 5);
    const int tilesN = DMc >> 6;                     // 8
    int mT = wave / tilesN;
    int n64 = wave - mT * tilesN;
    int Mtot = Bdim * lCount;
    if (mT * 16 >= Mtot) return;                     // wave-uniform
    int lo = lane & 15, hi = lane >> 4, k2 = hi * 2;

    int r = mT * 16 + lo;
    int bb = r / lCount;
    int ll = lBegin + (r - bb * lCount);
    const float* Arow = mem + ((size_t)bb * Lc + ll) * K + k2;
    const float* B0 = ctrlW + (size_t)(n64 * 64 + lo) * K + k2;
    const float* B1 = B0 + (size_t)16 * K;
    const float* B2 = B0 + (size_t)32 * K;
    const float* B3 = B0 + (size_t)48 * K;

    v8f c0 = (v8f)0.0f, c1 = (v8f)0.0f, c2 = (v8f)0.0f, c3 = (v8f)0.0f;

    v2f a0 = *(const v2f*)(Arow);
    v2f p0 = *(const v2f*)(B0);
    v2f p1 = *(const v2f*)(B1);
    v2f p2 = *(const v2f*)(B2);
    v2f p3 = *(const v2f*)(B3);
    v2f a1 = *(const v2f*)(Arow + 4);
    v2f q0 = *(const v2f*)(B0 + 4);
    v2f q1 = *(const v2f*)(B1 + 4);
    v2f q2 = *(const v2f*)(B2 + 4);
    v2f q3 = *(const v2f*)(B3 + 4);

    for (int k = 8; k + 8 <= K; k += 8) {
        if ((k & 127) == 8 && k + 512 < K) {        // wave-uniform prefetch hints
            __builtin_prefetch(Arow + k + 512, 0, 0);
            __builtin_prefetch(B0 + k + 512, 0, 0);
            __builtin_prefetch(B1 + k + 512, 0, 0);
            __builtin_prefetch(B2 + k + 512, 0, 0);
            __builtin_prefetch(B3 + k + 512, 0, 0);
        }
        v2f na  = *(const v2f*)(Arow + k);
        v2f nb0 = *(const v2f*)(B0 + k);
        v2f nb1 = *(const v2f*)(B1 + k);
        v2f nb2 = *(const v2f*)(B2 + k);
        v2f nb3 = *(const v2f*)(B3 + k);
        c0 = wmma4(a0, p0, c0);
        c1 = wmma4(a0, p1, c1);
        c2 = wmma4(a0, p2, c2);
        c3 = wmma4(a0, p3, c3);
        a0 = na; p0 = nb0; p1 = nb1; p2 = nb2; p3 = nb3;
        v2f ma  = *(const v2f*)(Arow + k + 4);
        v2f mb0 = *(const v2f*)(B0 + k + 4);
        v2f mb1 = *(const v2f*)(B1 + k + 4);
        v2f mb2 = *(const v2f*)(B2 + k + 4);
        v2f mb3 = *(const v2f*)(B3 + k + 4);
        c0 = wmma4(a1, q0, c0);
        c1 = wmma4(a1, q1, c1);
        c2 = wmma4(a1, q2, c2);
        c3 = wmma4(a1, q3, c3);
        a1 = ma; q0 = mb0; q1 = mb1; q2 = mb2; q3 = mb3;
    }
    c0 = wmma4(a0, p0, c0); c1 = wmma4(a0, p1, c1);
    c2 = wmma4(a0, p2, c2); c3 = wmma4(a0, p3, c3);
    c0 = wmma4(a1, q0, c0); c1 = wmma4(a1, q1, c1);
    c2 = wmma4(a1, q2, c2); c3 = wmma4(a1, q3, c3);

#pragma unroll
    for (int i = 0; i < 8; ++i) {
        int r2 = mT * 16 + hi * 8 + i;
        int b2i = r2 / lCount;
        int l2i = lBegin + (r2 - b2i * lCount);
        size_t rowOff = ((size_t)b2i * Lc + l2i) * DMc;
        int col = n64 * 64 + lo;
        encSlots[rowOff + col]      = c0[i] + ctrlb[col];
        encSlots[rowOff + col + 16] = c1[i] + ctrlb[col + 16];
        encSlots[rowOff + col + 32] = c2[i] + ctrlb[col + 32];
        encSlots[rowOff + col + 48] = c3[i] + ctrlb[col + 48];
    }
}

// ---------------------------------------------------------------------------
// x = concat(op_token, root_token, encSlots[b])  -> [B, 34, DM]
// ---------------------------------------------------------------------------
__global__ void build_x_kernel(const float* __restrict__ encSlots, const float* __restrict__ opTok,
                               const float* __restrict__ rootTok, float* __restrict__ x)
{
    int idx = blockIdx.x * blockDim.x + threadIdx.x;
    if (idx >= Bdim * Tc * DMc) return;
    int d = idx % DMc;
    int t = (idx / DMc) % Tc;
    int b = idx / (DMc * Tc);
    float v;
    if (t == 0)      v = opTok[d];
    else if (t == 1) v = rootTok[d];
    else             v = encSlots[((size_t)b * Lc + (t - 2)) * DMc + d];
    x[idx] = v;
}

// ---------------------------------------------------------------------------
// LayerNorm over DM=512, one block (128 thr) per row.
// ---------------------------------------------------------------------------
__global__ void layernorm_kernel(const float* __restrict__ X, const float* __restrict__ g,
                                 const float* __restrict__ bt, float* __restrict__ Y)
{
    __shared__ float red[128];
    int row = blockIdx.x, tid = threadIdx.x;
    const float* x = X + (size_t)row * DMc;
    float v0 = x[tid], v1 = x[tid + 128], v2 = x[tid + 256], v3 = x[tid + 384];
    red[tid] = v0 + v1 + v2 + v3;
    __syncthreads();
    for (int off = 64; off > 0; off >>= 1) { if (tid < off) red[tid] += red[tid + off]; __syncthreads(); }
    float mean = red[0] * (1.0f / DMc);
    __syncthreads();
    float d0 = v0 - mean, d1 = v1 - mean, d2 = v2 - mean, d3 = v3 - mean;
    red[tid] = d0 * d0 + d1 * d1 + d2 * d2 + d3 * d3;
    __syncthreads();
    for (int off = 64; off > 0; off >>= 1) { if (tid < off) red[tid] += red[tid + off]; __syncthreads(); }
    float inv = rsqrtf(red[0] * (1.0f / DMc) + 1e-5f);
    float* y = Y + (size_t)row * DMc;
    y[tid]       = d0 * inv * g[tid]       + bt[tid];
    y[tid + 128] = d1 * inv * g[tid + 128] + bt[tid + 128];
    y[tid + 256] = d2 * inv * g[tid + 256] + bt[tid + 256];
    y[tid + 384] = d3 * inv * g[tid + 384] + bt[tid + 384];
}

// ---------------------------------------------------------------------------
// MHA over 34 tokens; one block per (b, h). qkv = [B,34,3*DM].
// ---------------------------------------------------------------------------
__global__ void attention_kernel(const float* __restrict__ qkv, float* __restrict__ obuf)
{
    __shared__ float qs[Tc * DHc], ks[Tc * DHc], vs[Tc * DHc], sc[Tc * Tc];
    int b = blockIdx.x / NHc, h = blockIdx.x % NHc;
    const float* base = qkv + (size_t)b * Tc * 3 * DMc + h * DHc;
    for (int idx = threadIdx.x; idx < Tc * DHc; idx += blockDim.x) {
        int t = idx / DHc, d = idx - t * DHc;
        const float* row = base + (size_t)t * 3 * DMc;
        qs[idx] = row[d];
        ks[idx] = row[DMc + d];
        vs[idx] = row[2 * DMc + d];
    }
    __syncthreads();
    for (int p = threadIdx.x; p < Tc * Tc; p += blockDim.x) {
        int qi = p / Tc, kj = p - qi * Tc;
        float s = 0.0f;
#pragma unroll 8
        for (int d = 0; d < DHc; ++d) s += qs[qi * DHc + d] * ks[kj * DHc + d];
        sc[p] = s * 0.125f;                     // 1/sqrt(64)
    }
    __syncthreads();
    for (int qi = threadIdx.x; qi < Tc; qi += blockDim.x) {
        float mx = -3.4e38f;
        for (int kj = 0; kj < Tc; ++kj) mx = fmaxf(mx, sc[qi * Tc + kj]);
        float sum = 0.0f;
        for (int kj = 0; kj < Tc; ++kj) { float e = expf(sc[qi * Tc + kj] - mx); sc[qi * Tc + kj] = e; sum += e; }
        float inv = 1.0f / sum;
        for (int kj = 0; kj < Tc; ++kj) sc[qi * Tc + kj] *= inv;
    }
    __syncthreads();
    for (int idx = threadIdx.x; idx < Tc * DHc; idx += blockDim.x) {
        int qi = idx / DHc, d = idx - qi * DHc;
        float o = 0.0f;
        for (int kj = 0; kj < Tc; ++kj) o += sc[qi * Tc + kj] * vs[kj * DHc + d];
        obuf[((size_t)b * Tc + qi) * DMc + h * DHc + d] = o;
    }
}

// ---------------------------------------------------------------------------
// Heads: op softmax + root_f + arg softmax-over-slots.
// Folds op-dist into outputs: wcomb[b,l,n] = argsoftmax[l,n] * opw[n],
// rfScaled[b,f] = op2 * root_f[b,f].  One block (256 thr) per batch.
// ---------------------------------------------------------------------------
__device__ __forceinline__ float dot512(const float* __restrict__ a, const float* __restrict__ w)
{
    float s = 0.0f;
#pragma unroll 8
    for (int i = 0; i < DMc; ++i) s += a[i] * w[i];
    return s;
}

__global__ void heads_kernel(const float* __restrict__ xf,
                             const float* __restrict__ argW, const float* __restrict__ argB,
                             const float* __restrict__ rootfW, const float* __restrict__ rootfB,
                             const float* __restrict__ opW, const float* __restrict__ opB,
                             float* __restrict__ wcomb, float* __restrict__ rfScaled)
{
    __shared__ float opl[3];
    __shared__ float rootf[Fc];
    __shared__ float argl[Lc * 4];
    __shared__ float opp[4];
    int b = blockIdx.x, j = threadIdx.x;
    const float* xb = xf + (size_t)b * Tc * DMc;
    if (j < 3) {
        opl[j] = dot512(xb, opW + (size_t)j * DMc) + opB[j];
    } else if (j < 3 + Fc) {
        int f = j - 3;
        rootf[f] = dot512(xb + DMc, rootfW + (size_t)f * DMc) + rootfB[f];
    } else if (j < 3 + Fc + Lc * 4) {
        int q = j - (3 + Fc);
        int l = q >> 2, n = q & 3;
        argl[l * 4 + n] = dot512(xb + (size_t)(2 + l) * DMc, argW + (size_t)n * DMc) + argB[n];
    }
    __syncthreads();
    if (j == 0) {
        float mx = fmaxf(opl[0], fmaxf(opl[1], opl[2]));
        float e0 = expf(opl[0] - mx), e1 = expf(opl[1] - mx), e2 = expf(opl[2] - mx);
        float inv = 1.0f / (e0 + e1 + e2);
        opp[0] = e0 * inv; opp[1] = e1 * inv; opp[2] = e2 * inv; opp[3] = e2 * inv;
    }
    __syncthreads();
    if (j < 4) {                                    // softmax over slots, col j
        float mx = -3.4e38f;
        for (int l = 0; l < Lc; ++l) mx = fmaxf(mx, argl[l * 4 + j]);
        float sum = 0.0f;
        for (int l = 0; l < Lc; ++l) sum += expf(argl[l * 4 + j] - mx);
        float sc = opp[j] / sum;
        for (int l = 0; l < Lc; ++l)
            wcomb[(size_t)b * Lc * 4 + l * 4 + j] = expf(argl[l * 4 + j] - mx) * sc;
    } else if (j < 4 + Fc) {
        int f = j - 4;
        rfScaled[b * Fc + f] = opp[2] * rootf[f];
    }
}

// ---------------------------------------------------------------------------
// aScaled[b,n,:,:] = sum_l wcomb[b,l,n] * mem[b,l,:,:]   (L2-resident gather)
// ---------------------------------------------------------------------------
__global__ void interp_read_kernel(const float* __restrict__ mem, const float* __restrict__ wcomb,
                                   float* __restrict__ aScaled)
{
    __shared__ float wsh[Lc * 4];
    int b = blockIdx.x >> 5;                         // 32 blocks per batch
    int chunk = blockIdx.x & 31;
    int tid = threadIdx.x;
    if (tid < Lc * 4) wsh[tid] = wcomb[(size_t)b * Lc * 4 + tid];
    __syncthreads();
    int i4 = chunk * 256 + tid;                      // float4 index in [0, 8192)
    size_t off = (size_t)i4 * 4;
    float4 a0 = {0, 0, 0, 0}, a1 = {0, 0, 0, 0}, a2 = {0, 0, 0, 0}, a3 = {0, 0, 0, 0};
    const float* mb = mem + (size_t)b * Lc * FRc + off;
    for (int l = 0; l < Lc; ++l) {
        float4 m = *(const float4*)(mb + (size_t)l * FRc);
        float w0 = wsh[l * 4 + 0], w1 = wsh[l * 4 + 1], w2 = wsh[l * 4 + 2], w3 = wsh[l * 4 + 3];
        a0.x += w0 * m.x; a0.y += w0 * m.y; a0.z += w0 * m.z; a0.w += w0 * m.w;
        a1.x += w1 * m.x; a1.y += w1 * m.y; a1.z += w1 * m.z; a1.w += w1 * m.w;
        a2.x += w2 * m.x; a2.y += w2 * m.y; a2.z += w2 * m.z; a2.w += w2 * m.w;
        a3.x += w3 * m.x; a3.y += w3 * m.y; a3.z += w3 * m.z; a3.w += w3 * m.w;
    }
    float* ab = aScaled + (size_t)b * 4 * FRc + off;
    *(float4*)(ab)           = a0;
    *(float4*)(ab + FRc)     = a1;
    *(float4*)(ab + 2 * FRc) = a2;
    *(float4*)(ab + 3 * FRc) = a3;
}

// ---------------------------------------------------------------------------
// out[b,f,t] = sum_n aScaled[b,n] @ Wn^T + rfScaled[b,f]*root_role[t];
// writes memory slot `slot`. One wave = 16x64 tile; 32 waves per batch.
// Pipelined per 512-deep K segment (4 segments: D_l, D_r, E_l, E_r).
// ---------------------------------------------------------------------------
__global__ void interp_out_kernel(const float* __restrict__ aScaled,
                                  const float* __restrict__ Dl, const float* __restrict__ Dr,
                                  const float* __restrict__ El, const float* __restrict__ Er,
                                  const float* __restrict__ rfScaled, const float* __restrict__ rootRole,
                                  float* __restrict__ memOut, int slot)
{
    int lane = threadIdx.x & 31;
    int wave = blockIdx.x * (blockDim.x >> 5) + (threadIdx.x >> 5);
    int b  = wave >> 5;
    int w2 = wave & 31;
    int mT = w2 >> 3;
    int n64 = w2 & 7;
    int lo = lane & 15, hi = lane >> 4, k2 = hi * 2;

    const float* Ws[4] = {Dl, Dr, El, Er};
    v8f c0 = (v8f)0.0f, c1 = (v8f)0.0f, c2 = (v8f)0.0f, c3 = (v8f)0.0f;
    for (int n = 0; n < 4; ++n) {
        const float* Arow = aScaled + ((size_t)(b * 4 + n) * Fc + mT * 16 + lo) * Rc + k2;
        const float* B0 = Ws[n] + (size_t)(n64 * 64 + lo) * Rc + k2;
        const float* B1 = B0 + (size_t)16 * Rc;
        const float* B2 = B0 + (size_t)32 * Rc;
        const float* B3 = B0 + (size_t)48 * Rc;

        v2f a0 = *(const v2f*)(Arow);
        v2f p0 = *(const v2f*)(B0);
        v2f p1 = *(const v2f*)(B1);
        v2f p2 = *(const v2f*)(B2);
        v2f p3 = *(const v2f*)(B3);
        v2f a1 = *(const v2f*)(Arow + 4);
        v2f q0 = *(const v2f*)(B0 + 4);
        v2f q1 = *(const v2f*)(B1 + 4);
        v2f q2 = *(const v2f*)(B2 + 4);
        v2f q3 = *(const v2f*)(B3 + 4);

        for (int k = 8; k + 8 <= Rc; k += 8) {
            v2f na  = *(const v2f*)(Arow + k);
            v2f nb0 = *(const v2f*)(B0 + k);
            v2f nb1 = *(const v2f*)(B1 + k);
            v2f nb2 = *(const v2f*)(B2 + k);
            v2f nb3 = *(const v2f*)(B3 + k);
            c0 = wmma4(a0, p0, c0);
            c1 = wmma4(a0, p1, c1);
            c2 = wmma4(a0, p2, c2);
            c3 = wmma4(a0, p3, c3);
            a0 = na; p0 = nb0; p1 = nb1; p2 = nb2; p3 = nb3;
            v2f ma  = *(const v2f*)(Arow + k + 4);
            v2f mb0 = *(const v2f*)(B0 + k + 4);
            v2f mb1 = *(const v2f*)(B1 + k + 4);
            v2f mb2 = *(const v2f*)(B2 + k + 4);
            v2f mb3 = *(const v2f*)(B3 + k + 4);
            c0 = wmma4(a1, q0, c0);
            c1 = wmma4(a1, q1, c1);
            c2 = wmma4(a1, q2, c2);
            c3 = wmma4(a1, q3, c3);
            a1 = ma; q0 = mb0; q1 = mb1; q2 = mb2; q3 = mb3;
        }
        c0 = wmma4(a0, p0, c0); c1 = wmma4(a0, p1, c1);
        c2 = wmma4(a0, p2, c2); c3 = wmma4(a0, p3, c3);
        c0 = wmma4(a1, q0, c0); c1 = wmma4(a1, q1, c1);
        c2 = wmma4(a1, q2, c2); c3 = wmma4(a1, q3, c3);
    }
#pragma unroll
    for (int i = 0; i < 8; ++i) {
        int f = mT * 16 + hi * 8 + i;
        float rf = rfScaled[b * Fc + f];
        int col = n64 * 64 + lo;
        size_t rowOff = (((size_t)b * Lc + slot) * Fc + f) * Rc;
        memOut[rowOff + col]      = c0[i] + rf * rootRole[col];
        memOut[rowOff + col + 16] = c1[i] + rf * rootRole[col + 16];
        memOut[rowOff + col + 32] = c2[i] + rf * rootRole[col + 32];
        memOut[rowOff + col + 48] = c3[i] + rf * rootRole[col + 48];
    }
}

// ---------------------------------------------------------------------------
// Host orchestration
// ---------------------------------------------------------------------------
static void launch_gemm(const float* X, const float* W, const float* bias, const float* residual,
                        float* Y, int M, int N, int K, int act, hipStream_t stream)
{
    int waves = (M / 16) * (N / 64);
    int blocks = (waves + 3) / 4;
    gemm_xwT_kernel<<<blocks, 128, 0, stream>>>(X, W, bias, residual, Y, M, N, K, act);
}

extern "C" void kernel_launch(void* const* d_in, const int* in_sizes, int n_in,
                              void* d_out, int out_size, void* d_ws, size_t ws_size,
                              hipStream_t stream)
{
    (void)in_sizes; (void)n_in; (void)out_size; (void)ws_size;
    const float* memIn    = (const float*)d_in[0];
    const float* ctrlW    = (const float*)d_in[1];
    const float* ctrlb    = (const float*)d_in[2];
    const float* opTok    = (const float*)d_in[3];
    const float* rootTok  = (const float*)d_in[4];
    const float* ln1g     = (const float*)d_in[5];
    const float* ln1b     = (const float*)d_in[6];
    const float* Wqkv     = (const float*)d_in[7];
    const float* bqkv     = (const float*)d_in[8];
    const float* Wo       = (const float*)d_in[9];
    const float* bo       = (const float*)d_in[10];
    const float* ln2g     = (const float*)d_in[11];
    const float* ln2b     = (const float*)d_in[12];
    const float* W1       = (const float*)d_in[13];
    const float* b1       = (const float*)d_in[14];
    const float* W2       = (const float*)d_in[15];
    const float* b2       = (const float*)d_in[16];
    const float* lnfg     = (const float*)d_in[17];
    const float* lnfb     = (const float*)d_in[18];
    const float* argW     = (const float*)d_in[19];
    const float* argB     = (const float*)d_in[20];
    const float* rootfW   = (const float*)d_in[21];
    const float* rootfB   = (const float*)d_in[22];
    const float* opW      = (const float*)d_in[23];
    const float* opB      = (const float*)d_in[24];
    const float* Dl       = (const float*)d_in[25];
    const float* Dr       = (const float*)d_in[26];
    const float* El       = (const float*)d_in[27];
    const float* Er       = (const float*)d_in[28];
    const float* rootRole = (const float*)d_in[29];

    float* memCur = (float*)d_out;                   // evolving memory lives in d_out
    float* ws = (float*)d_ws;
    const size_t SZ_ENC = (size_t)Bdim * Lc * DMc;        // 524288
    const size_t SZ_X   = (size_t)Bdim * Tc * DMc;        // 557056
    const size_t SZ_QKV = (size_t)Bdim * Tc * 3 * DMc;    // 1671168
    const size_t SZ_FFN = (size_t)Bdim * Tc * DFFc;       // 2228224
    float* encSlots = ws;                 ws += SZ_ENC;
    float* xbuf     = ws;                 ws += SZ_X;
    float* hbuf     = ws;                 ws += SZ_X;
    float* qkvbuf   = ws;                 ws += SZ_QKV;
    float* obuf     = ws;                 ws += SZ_X;
    float* ffnbuf   = ws;                 ws += SZ_FFN;
    float* wcomb    = ws;                 ws += (size_t)Bdim * Lc * 4;
    float* rfScaled = ws;                 ws += (size_t)Bdim * Fc;
    float* aScaled  = ws;                 /* Bdim*4*FRc */

    // 1) memory -> d_out (slots beyond L0 arrive zeroed from setup)
    hipMemcpyAsync(memCur, memIn, sizeof(float) * (size_t)Bdim * Lc * FRc,
                   hipMemcpyDeviceToDevice, stream);

    // 2) one-time full slot encoding: (B*L=1024) x 32768 x 512 GEMM
    {
        int waves = (Bdim * Lc / 16) * (DMc / 64);   // 512
        encode_slots_kernel<<<(waves + 3) / 4, 128, 0, stream>>>(memCur, ctrlW, ctrlb, encSlots, 0, Lc);
    }

    const int M = Bdim * Tc;                         // 1088 token rows
    for (int s = 0; s < NSTEPS; ++s) {
        build_x_kernel<<<(int)((SZ_X + 255) / 256), 256, 0, stream>>>(encSlots, opTok, rootTok, xbuf);
        layernorm_kernel<<<M, 128, 0, stream>>>(xbuf, ln1g + (size_t)s * DMc, ln1b + (size_t)s * DMc, hbuf);
        launch_gemm(hbuf, Wqkv + (size_t)s * 3 * DMc * DMc, bqkv + (size_t)s * 3 * DMc,
                    nullptr, qkvbuf, M, 3 * DMc, DMc, 0, stream);
        attention_kernel<<<Bdim * NHc, 128, 0, stream>>>(qkvbuf, obuf);
        launch_gemm(obuf, Wo + (size_t)s * DMc * DMc, bo + (size_t)s * DMc,
                    xbuf, xbuf, M, DMc, DMc, 0, stream);                       // x += o@Wo^T+bo
        layernorm_kernel<<<M, 128, 0, stream>>>(xbuf, ln2g + (size_t)s * DMc, ln2b + (size_t)s * DMc, hbuf);
        launch_gemm(hbuf, W1 + (size_t)s * DFFc * DMc, b1 + (size_t)s * DFFc,
                    nullptr, ffnbuf, M, DFFc, DMc, 1, stream);                 // gelu
        launch_gemm(ffnbuf, W2 + (size_t)s * DMc * DFFc, b2 + (size_t)s * DMc,
                    xbuf, xbuf, M, DMc, DFFc, 0, stream);                      // x += ffn@W2^T+b2
        layernorm_kernel<<<M, 128, 0, stream>>>(xbuf, lnfg + (size_t)s * DMc, lnfb + (size_t)s * DMc, hbuf);
        heads_kernel<<<Bdim, 256, 0, stream>>>(hbuf, argW, argB, rootfW, rootfB, opW, opB,
                                               wcomb, rfScaled);
        interp_read_kernel<<<Bdim * 32, 256, 0, stream>>>(memCur, wcomb, aScaled);
        interp_out_kernel<<<(Bdim * 32) / 4, 128, 0, stream>>>(aScaled, Dl, Dr, El, Er,
                                                               rfScaled, rootRole, memCur, L0c + s);
        // incremental re-encode of the freshly written slot (M = 32 rows)
        {
            int waves = (Bdim / 16) * (DMc / 64);    // 16
            encode_slots_kernel<<<(waves + 3) / 4, 128, 0, stream>>>(memCur, ctrlW, ctrlb, encSlots,
                                                                     L0c + s, 1);
        }
    }
}